// KroneckerHamiltonianModelV2_11020886081998
// MI455X (gfx1250) — compile-verified
//
#include <hip/hip_runtime.h>
#include <hip/hip_bf16.h>
#include <math.h>

#define N_NODES 2048
#define NA_H 1024
#define E_EDGES 131072
#define MUL 64
#define BD 4
#define KKEEP 20
#define KS 8
#define INV_SQRT3 0.57735026918962576f

typedef float v2f __attribute__((ext_vector_type(2)));
typedef float v8f __attribute__((ext_vector_type(8)));
typedef unsigned int v4u __attribute__((ext_vector_type(4)));
typedef int v4i __attribute__((ext_vector_type(4)));
typedef int v8i __attribute__((ext_vector_type(8)));

__device__ __forceinline__ float silu_f(float x) { return x / (1.0f + __expf(-x)); }

__device__ __forceinline__ v8f wmma_f32_16x16x4(v2f a, v2f b, v8f c) {
    // V_WMMA_F32_16X16X4_F32 : D(16x16,f32) = A(16x4,f32) x B(4x16,f32) + C
    return __builtin_amdgcn_wmma_f32_16x16x4_f32(false, a, false, b, (short)0, c, false, false);
}

// TDM: DMA a dense [rows x cols] f32 tile from global into LDS (tensor == tile,
// contiguous). D# layout per CDNA5 ISA.
__device__ __forceinline__ void tdm_load_2d_f32(const float* gptr, unsigned lds_byte_addr,
                                                int cols, int rows) {
    unsigned long long ga = (unsigned long long)(__SIZE_TYPE__)gptr;
    v4u g0;
    g0.x = 1u;                                        // count=1, is_restore=0, gather off
    g0.y = lds_byte_addr;                             // D#.lds_addr
    g0.z = (unsigned)(ga & 0xffffffffu);              // global_addr[31:0]
    g0.w = (unsigned)(ga >> 32) | (2u << 30);         // global_addr[56:32] | type=2
    v8i g1;
    g1[0] = (2 << 16);                                // data_size=2 (4 bytes)
    g1[1] = (cols << 16);                             // tensor_dim0 low16
    g1[2] = (rows << 16);                             // tensor_dim0 hi=0, tensor_dim1 lo=rows
    g1[3] = (cols << 16);                             // tensor_dim1 hi=0, tile_dim0=cols
    g1[4] = rows;                                     // tile_dim1=rows, tile_dim2=0
    g1[5] = cols;                                     // tensor_dim0_stride (low32)
    g1[6] = 0;
    g1[7] = 0;
    v4i g2 = {0, 0, 0, 0};
    v4i g3 = {0, 0, 0, 0};
#if __clang_major__ >= 23
    v8i g4 = {0, 0, 0, 0, 0, 0, 0, 0};
    __builtin_amdgcn_tensor_load_to_lds(g0, g1, g2, g3, g4, 0);
#else
    __builtin_amdgcn_tensor_load_to_lds(g0, g1, g2, g3, 0);
#endif
}

// ---------------------------------------------------------------------------
// Re-pack a [K x Ncols] row-major weight into K-pair-major v2f layout:
// out[k2*Ncols + n] = { B[(2k2)*Ncols + n], B[(2k2+1)*Ncols + n] }
// so a WMMA B-fragment load is a single b64.
__global__ void pair_kernel(const float* __restrict__ B, v2f* __restrict__ out,
                            int K, int Ncols) {
    int t = blockIdx.x * 256 + threadIdx.x;
    int total = (K / 2) * Ncols;
    if (t >= total) return;
    int k2 = t / Ncols, n = t - k2 * Ncols;
    v2f p;
    p.x = B[(2 * k2) * Ncols + n];
    p.y = B[(2 * k2 + 1) * Ncols + n];
    out[t] = p;
}

// s[n][u] = embed[z[n]][u]
__global__ void embed_kernel(const int* __restrict__ z, const float* __restrict__ embed,
                             float* __restrict__ s) {
    int t = blockIdx.x * 256 + threadIdx.x;
    if (t >= N_NODES * MUL) return;
    int n = t >> 6, u = t & 63;
    s[t] = embed[z[n] * MUL + u];
}

// dist[e] = |vec|, sh1[e] = sqrt(3)*vec/dist
__global__ void edge_geom_kernel(const float* __restrict__ vec, float* __restrict__ dist,
                                 float* __restrict__ sh1) {
    int e = blockIdx.x * 256 + threadIdx.x;
    if (e >= E_EDGES) return;
    float x = vec[e * 3 + 0], y = vec[e * 3 + 1], zc = vec[e * 3 + 2];
    float d = sqrtf(x * x + y * y + zc * zc);
    dist[e] = d;
    float sc = 1.7320508075688772f / d;
    sh1[e * 3 + 0] = x * sc;
    sh1[e * 3 + 1] = y * sc;
    sh1[e * 3 + 2] = zc * sc;
}

// ---------------------------------------------------------------------------
// Fused per-edge: h1 = silu(dist*fc1) (rank-1, kept in registers as WMMA A
// fragments); h2 = silu(h1@fc2) with fc2 staged into LDS by the Tensor Data
// Mover; w = h2@fc3 (paired b64 loads) consumed immediately as TP messages
// scattered with f32 atomics. 4 waves/block, 16 edges/wave.
__global__ void __launch_bounds__(128) edge_layer_kernel(
    const int* __restrict__ src, const int* __restrict__ dst,
    const float* __restrict__ dist, const float* __restrict__ sh1,
    const float* __restrict__ s, const float* __restrict__ v,
    const float* __restrict__ fc1, const v2f* __restrict__ fc2p,
    const v2f* __restrict__ fc3p,
    float* __restrict__ agg_s, float* __restrict__ agg_v) {
    __shared__ float h2[4][16][68];
    __shared__ v2f wfc2[32 * 64]; // paired fc2, 16KB
    const int wave = threadIdx.x >> 5;
    const int lane = threadIdx.x & 31;
    const int m = lane & 15;    // A row / B-D col within tile
    const int half = lane >> 4; // K sub-select
    const int e0 = blockIdx.x * 64 + wave * 16;

    // async stage paired fc2 (16KB) into LDS via TDM; overlaps with h1 frags
    if (wave == 0) {
        tdm_load_2d_f32((const float*)fc2p, (unsigned)(__SIZE_TYPE__)(&wfc2[0]), 128, 32);
    }
    __builtin_prefetch(fc3p);

    // A fragments of h1 = silu(dist * fc1): lane (m, half) needs columns
    // {4i+2*half, 4i+2*half+1}; rank-1 so computed locally, no LDS.
    const float dm = dist[e0 + m];
    v2f afrag[8];
#pragma unroll
    for (int i = 0; i < 8; ++i) {
        afrag[i].x = silu_f(dm * fc1[4 * i + 2 * half]);
        afrag[i].y = silu_f(dm * fc1[4 * i + 2 * half + 1]);
    }
    if (wave == 0) __builtin_amdgcn_s_wait_tensorcnt(0);
    __syncthreads();

    // h2 = silu(h1 @ fc2)   [16x64] = [16x64][64x64], B paired from LDS
#pragma unroll
    for (int ct = 0; ct < 4; ++ct) {
        v8f acc = {0.f, 0.f, 0.f, 0.f, 0.f, 0.f, 0.f, 0.f};
#pragma unroll
        for (int k0 = 0; k0 < 64; k0 += 4) {
            v2f b = wfc2[(k0 / 2 + half) * 64 + ct * 16 + m];
            acc = wmma_f32_16x16x4(afrag[k0 / 4], b, acc);
        }
#pragma unroll
        for (int r = 0; r < 8; ++r)
            h2[wave][r + 8 * half][ct * 16 + m] = silu_f(acc[r]);
    }
    __syncthreads();

    // w = h2 @ fc3  [16x256]; consume each 16x16 tile immediately as messages
    for (int ct = 0; ct < 16; ++ct) {
        const int path = ct >> 2;
        const int u = (ct & 3) * 16 + m;
        v8f acc = {0.f, 0.f, 0.f, 0.f, 0.f, 0.f, 0.f, 0.f};
#pragma unroll
        for (int k0 = 0; k0 < 64; k0 += 4) {
            v2f a, b;
            a.x = h2[wave][m][k0 + 2 * half];
            a.y = h2[wave][m][k0 + 2 * half + 1];
            b = fc3p[(k0 / 2 + half) * 256 + ct * 16 + m];
            acc = wmma_f32_16x16x4(a, b, acc);
        }
#pragma unroll
        for (int r = 0; r < 8; ++r) {
            const int e = e0 + r + 8 * half;
            const float wv = acc[r];
            const int sn = src[e];
            const int dn = dst[e];
            if (path == 0) { // 0e x Y0 -> 0e
                atomicAdd(&agg_s[dn * MUL + u], wv * s[sn * MUL + u]);
            } else if (path == 1) { // 0e x Y1 -> 1o
                float ss = s[sn * MUL + u];
#pragma unroll
                for (int c = 0; c < 3; ++c)
                    atomicAdd(&agg_v[(dn * MUL + u) * 3 + c], wv * ss * sh1[e * 3 + c]);
            } else if (path == 2) { // 1o x Y0 -> 1o
#pragma unroll
                for (int c = 0; c < 3; ++c)
                    atomicAdd(&agg_v[(dn * MUL + u) * 3 + c], wv * v[(sn * MUL + u) * 3 + c]);
            } else { // 1o x Y1 -> 0e
                float dotv = 0.f;
#pragma unroll
                for (int c = 0; c < 3; ++c)
                    dotv += v[(sn * MUL + u) * 3 + c] * sh1[e * 3 + c];
                atomicAdd(&agg_s[dn * MUL + u], wv * INV_SQRT3 * dotv);
            }
        }
    }
}

// ---------------------------------------------------------------------------
// Generic [Mx64] @ [64x64] WMMA GEMM (B pre-paired) with strided A/D, optional
// residual, bias and silu. 4 waves per block, one 16x16 tile per wave.
__global__ void __launch_bounds__(128) gemm64_kernel(
    const float* __restrict__ A, int lda, int sa,
    const v2f* __restrict__ Bp,
    const float* __restrict__ resid, int ldr, int sr,
    const float* __restrict__ bias, int act,
    float* __restrict__ D, int ldd, int sd) {
    const int wave = threadIdx.x >> 5;
    const int lane = threadIdx.x & 31;
    const int m = lane & 15;
    const int half = lane >> 4;
    const int m0 = blockIdx.x * 16;
    const int n0 = wave * 16;
    v8f acc = {0.f, 0.f, 0.f, 0.f, 0.f, 0.f, 0.f, 0.f};
#pragma unroll
    for (int k0 = 0; k0 < 64; k0 += 4) {
        v2f a, b;
        a.x = A[(m0 + m) * lda + (k0 + 2 * half) * sa];
        a.y = A[(m0 + m) * lda + (k0 + 2 * half + 1) * sa];
        b = Bp[(k0 / 2 + half) * 64 + n0 + m];
        acc = wmma_f32_16x16x4(a, b, acc);
    }
#pragma unroll
    for (int r = 0; r < 8; ++r) {
        const int row = m0 + r + 8 * half;
        const int col = n0 + m;
        float x = acc[r];
        if (resid) x += resid[row * ldr + col * sr];
        if (bias) x += bias[col];
        if (act) x = silu_f(x);
        D[row * ldd + col * sd] = x;
    }
}

// onsite = t1 @ ons_w2 + b2   [Nx64][64x4]
__global__ void head4_kernel(const float* __restrict__ t1, const float* __restrict__ w2,
                             const float* __restrict__ b2, float* __restrict__ onsite) {
    int t = blockIdx.x * 256 + threadIdx.x;
    if (t >= N_NODES * BD) return;
    int n = t >> 2, b = t & 3;
    float acc = b2[b];
    for (int j = 0; j < 64; ++j) acc += t1[n * 64 + j] * w2[j * 4 + b];
    onsite[t] = acc;
}

// U[p][k] = a[n,k,b]: low-rank factor columns (p = 4*local_n + b)
__global__ void lowrank_kernel(const float* __restrict__ s, const float* __restrict__ v,
                               const float* __restrict__ WsA, const float* __restrict__ WvA,
                               const float* __restrict__ WsB, const float* __restrict__ WvB,
                               float* __restrict__ UA, float* __restrict__ UB) {
    int t = blockIdx.x * 256 + threadIdx.x;
    if (t >= N_NODES * 4) return;
    int n = t >> 2, k = t & 3;
    const float* Ws = (n < NA_H) ? WsA : WsB;
    const float* Wv = (n < NA_H) ? WvA : WvB;
    float as = 0.f, a0 = 0.f, a1 = 0.f, a2 = 0.f;
    for (int u = 0; u < 64; ++u) {
        float wsv = Ws[k * 64 + u], wvv = Wv[k * 64 + u];
        as += s[n * 64 + u] * wsv;
        a0 += v[(n * 64 + u) * 3 + 0] * wvv;
        a1 += v[(n * 64 + u) * 3 + 1] * wvv;
        a2 += v[(n * 64 + u) * 3 + 2] * wvv;
    }
    float* U = (n < NA_H) ? UA : UB;
    int ln = (n < NA_H) ? n : n - NA_H;
    U[(ln * 4 + 0) * 4 + k] = as;
    U[(ln * 4 + 1) * 4 + k] = a0;
    U[(ln * 4 + 2) * 4 + k] = a1;
    U[(ln * 4 + 3) * 4 + k] = a2;
}

// ---------------------------------------------------------------------------
// Eigenvalues of H = diag(d) + U U^T (4096x4096, rank-4 update) by bisection
// on the inertia counting function N(l) = #{d<l} - #{eig(I + U^T(D-l)^-1 U)<=0}.
__device__ __forceinline__ float block_reduce_op(float x, float* red, int op) {
    int tid = threadIdx.x;
    red[tid] = x;
    __syncthreads();
    for (int st = 128; st > 0; st >>= 1) {
        if (tid < st) {
            float a = red[tid], b = red[tid + st];
            red[tid] = (op == 0) ? (a + b) : (op == 1 ? fminf(a, b) : fmaxf(a, b));
        }
        __syncthreads();
    }
    float r = red[0];
    __syncthreads();
    return r;
}

__global__ void __launch_bounds__(256) eigs_kernel(const float* __restrict__ onsite,
                                                   const float* __restrict__ UA,
                                                   const float* __restrict__ UB,
                                                   float* __restrict__ evA, float* __restrict__ evB) {
    __shared__ float red[256];
    __shared__ float s_lo, s_hi;
    const int half = blockIdx.x / KKEEP;
    const int kth = (blockIdx.x % KKEEP) + 1;
    const float* d = onsite + half * NA_H * BD; // 4096 diagonal entries
    const float* U = half ? UB : UA;
    float* ev = half ? evB : evA;
    const int tid = threadIdx.x;
    const int NP = NA_H * BD;

    // spectrum bracket: [min d - 1, max d + tr(UU^T) + 1]
    float dmin = 3.4e38f, dmax = -3.4e38f, ssq = 0.f;
    for (int p = tid; p < NP; p += 256) {
        float dp = d[p];
        dmin = fminf(dmin, dp);
        dmax = fmaxf(dmax, dp);
        const float* up = U + p * 4;
        ssq += up[0] * up[0] + up[1] * up[1] + up[2] * up[2] + up[3] * up[3];
    }
    dmin = block_reduce_op(dmin, red, 1);
    dmax = block_reduce_op(dmax, red, 2);
    ssq = block_reduce_op(ssq, red, 0);
    if (tid == 0) { s_lo = dmin - 1.0f; s_hi = dmax + ssq + 1.0f; }
    __syncthreads();

    for (int it = 0; it < 48; ++it) {
        float mid = 0.5f * (s_lo + s_hi);
        float cntf = 0.f;
        float S[10] = {0.f, 0.f, 0.f, 0.f, 0.f, 0.f, 0.f, 0.f, 0.f, 0.f};
        for (int p = tid; p < NP; p += 256) {
            float dp = d[p];
            cntf += (dp < mid) ? 1.f : 0.f;
            float r = 1.0f / (dp - mid);
            const float* up = U + p * 4;
            float u0 = up[0], u1 = up[1], u2 = up[2], u3 = up[3];
            S[0] += u0 * u0 * r; S[1] += u0 * u1 * r; S[2] += u0 * u2 * r; S[3] += u0 * u3 * r;
            S[4] += u1 * u1 * r; S[5] += u1 * u2 * r; S[6] += u1 * u3 * r;
            S[7] += u2 * u2 * r; S[8] += u2 * u3 * r; S[9] += u3 * u3 * r;
        }
        cntf = block_reduce_op(cntf, red, 0);
        for (int q = 0; q < 10; ++q) S[q] = block_reduce_op(S[q], red, 0);
        if (tid == 0) {
            float G[4][4] = {{1.f + S[0], S[1], S[2], S[3]},
                             {S[1], 1.f + S[4], S[5], S[6]},
                             {S[2], S[5], 1.f + S[7], S[8]},
                             {S[3], S[6], S[8], 1.f + S[9]}};
            const int P[6] = {0, 0, 0, 1, 1, 2};
            const int Q[6] = {1, 2, 3, 2, 3, 3};
            for (int sw = 0; sw < 10; ++sw) {
                for (int pr = 0; pr < 6; ++pr) {
                    int p = P[pr], q = Q[pr];
                    float apq = G[p][q];
                    if (fabsf(apq) > 1e-12f) {
                        float theta = 0.5f * (G[q][q] - G[p][p]) / apq;
                        float tt = ((theta >= 0.f) ? 1.f : -1.f) /
                                   (fabsf(theta) + sqrtf(theta * theta + 1.f));
                        float c = 1.f / sqrtf(tt * tt + 1.f), sn = tt * c;
                        for (int i = 0; i < 4; ++i) {
                            float gip = G[i][p], giq = G[i][q];
                            G[i][p] = c * gip - sn * giq;
                            G[i][q] = sn * gip + c * giq;
                        }
                        for (int i = 0; i < 4; ++i) {
                            float gpi = G[p][i], gqi = G[q][i];
                            G[p][i] = c * gpi - sn * gqi;
                            G[q][i] = sn * gpi + c * gqi;
                        }
                    }
                }
            }
            int nonpos = 0;
            for (int i = 0; i < 4; ++i) nonpos += (G[i][i] <= 0.f) ? 1 : 0;
            int Nmid = (int)(cntf + 0.5f) - nonpos;
            if (Nmid >= kth) s_hi = mid; else s_lo = mid;
        }
        __syncthreads();
    }
    if (tid == 0) ev[kth - 1] = 0.5f * (s_lo + s_hi);
}

// 20 smallest Kronecker sums + spectral features + final MLP -> scalar
__global__ void final_kernel(const float* __restrict__ evA, const float* __restrict__ evB,
                             const float* __restrict__ w1, const float* __restrict__ b1,
                             const float* __restrict__ w2, const float* __restrict__ b2,
                             float* __restrict__ out) {
    if (threadIdx.x != 0 || blockIdx.x != 0) return;
    float best[KKEEP];
    for (int i = 0; i < KKEEP; ++i) best[i] = 3.4e38f;
    for (int i = 0; i < KKEEP; ++i)
        for (int j = 0; j < KKEEP; ++j) {
            float sum = evA[i] + evB[j];
            if (sum < best[KKEEP - 1]) {
                int pos = KKEEP - 1;
                while (pos > 0 && best[pos - 1] > sum) { best[pos] = best[pos - 1]; --pos; }
                best[pos] = sum;
            }
        }
    float feat[3 * KS + 4];
    for (int q = 0; q < KS; ++q) {
        feat[q] = best[q];
        feat[KS + q] = evA[q];
        feat[2 * KS + q] = evB[q];
    }
    float gmin = 3.4e38f, gmax = -3.4e38f, sabs = 0.f, ssum = 0.f, ssq = 0.f;
    for (int i = 0; i < KS; ++i)
        for (int j = 0; j < KS; ++j) {
            float g = evA[i] - evB[j];
            gmin = fminf(gmin, g);
            gmax = fmaxf(gmax, g);
            sabs += fabsf(g);
            ssum += g;
            ssq += g * g;
        }
    const float cnt = (float)(KS * KS);
    float mu = ssum / cnt;
    float var = (ssq - cnt * mu * mu) / (cnt - 1.f);
    feat[24] = gmin;
    feat[25] = gmax;
    feat[26] = sabs / cnt;
    feat[27] = sqrtf(fmaxf(var, 0.f));
    float h[64];
    for (int j = 0; j < 64; ++j) {
        float acc = b1[j];
        for (int i = 0; i < 28; ++i) acc += feat[i] * w1[i * 64 + j];
        h[j] = silu_f(acc);
    }
    float o = b2[0];
    for (int j = 0; j < 64; ++j) o += h[j] * w2[j];
    out[0] = o;
}

// ---------------------------------------------------------------------------
extern "C" void kernel_launch(void* const* d_in, const int* in_sizes, int n_in,
                              void* d_out, int out_size, void* d_ws, size_t ws_size,
                              hipStream_t stream) {
    (void)in_sizes; (void)n_in; (void)out_size; (void)ws_size;
    const int* z = (const int*)d_in[0];
    const int* edge_index = (const int*)d_in[1];
    const float* edge_vec = (const float*)d_in[2];
    const float* embed = (const float*)d_in[3];
    const float *fc1[3], *fc2[3], *fc3[3], *lin_s[3], *lin_v[3];
    for (int l = 0; l < 3; ++l) {
        int base = 4 + l * 5;
        fc1[l] = (const float*)d_in[base + 0];
        fc2[l] = (const float*)d_in[base + 1];
        fc3[l] = (const float*)d_in[base + 2];
        lin_s[l] = (const float*)d_in[base + 3];
        lin_v[l] = (const float*)d_in[base + 4];
    }
    const float* ons_ext = (const float*)d_in[19];
    const float* ons_w1 = (const float*)d_in[20];
    const float* ons_b1 = (const float*)d_in[21];
    const float* ons_w2 = (const float*)d_in[22];
    const float* ons_b2 = (const float*)d_in[23];
    const float* vecA_ws = (const float*)d_in[24];
    const float* vecA_wv = (const float*)d_in[25];
    const float* vecB_ws = (const float*)d_in[26];
    const float* vecB_wv = (const float*)d_in[27];
    const float* en_w1 = (const float*)d_in[28];
    const float* en_b1 = (const float*)d_in[29];
    const float* en_w2 = (const float*)d_in[30];
    const float* en_b2 = (const float*)d_in[31];

    float* ws = (float*)d_ws;
    float* s = ws;       ws += N_NODES * MUL;
    float* v = ws;       ws += N_NODES * MUL * 3;
    float* dist = ws;    ws += E_EDGES;
    float* sh1 = ws;     ws += E_EDGES * 3;
    float* agg_s = ws;   ws += N_NODES * MUL;
    float* agg_v = ws;   ws += N_NODES * MUL * 3;
    float* sc = ws;      ws += N_NODES * MUL;
    float* t1 = ws;      ws += N_NODES * MUL;
    float* onsite = ws;  ws += N_NODES * BD;
    float* UA = ws;      ws += NA_H * BD * 4;
    float* UB = ws;      ws += NA_H * BD * 4;
    float* evA = ws;     ws += 32;
    float* evB = ws;     ws += 32;
    // paired weights (v2f elements, 2 floats each)
    v2f* fc2p[3];
    v2f* fc3p[3];
    v2f* lin_sp[3];
    v2f* lin_vp[3];
    for (int l = 0; l < 3; ++l) { fc2p[l] = (v2f*)ws; ws += 32 * 64 * 2; }
    for (int l = 0; l < 3; ++l) { fc3p[l] = (v2f*)ws; ws += 32 * 256 * 2; }
    for (int l = 0; l < 3; ++l) { lin_sp[l] = (v2f*)ws; ws += 32 * 64 * 2; }
    for (int l = 0; l < 3; ++l) { lin_vp[l] = (v2f*)ws; ws += 32 * 64 * 2; }
    v2f* ons_extp = (v2f*)ws; ws += 32 * 64 * 2;
    v2f* ons_w1p = (v2f*)ws;  ws += 32 * 64 * 2;

    const int* src = edge_index;
    const int* dst = edge_index + E_EDGES;

    // weight re-pack for b64 WMMA B-fragment loads
    for (int l = 0; l < 3; ++l) {
        pair_kernel<<<(32 * 64 + 255) / 256, 256, 0, stream>>>(fc2[l], fc2p[l], 64, 64);
        pair_kernel<<<(32 * 256 + 255) / 256, 256, 0, stream>>>(fc3[l], fc3p[l], 64, 256);
        pair_kernel<<<(32 * 64 + 255) / 256, 256, 0, stream>>>(lin_s[l], lin_sp[l], 64, 64);
        pair_kernel<<<(32 * 64 + 255) / 256, 256, 0, stream>>>(lin_v[l], lin_vp[l], 64, 64);
    }
    pair_kernel<<<(32 * 64 + 255) / 256, 256, 0, stream>>>(ons_ext, ons_extp, 64, 64);
    pair_kernel<<<(32 * 64 + 255) / 256, 256, 0, stream>>>(ons_w1, ons_w1p, 64, 64);

    embed_kernel<<<N_NODES * MUL / 256, 256, 0, stream>>>(z, embed, s);
    hipMemsetAsync(v, 0, (size_t)N_NODES * MUL * 3 * sizeof(float), stream);
    edge_geom_kernel<<<E_EDGES / 256, 256, 0, stream>>>(edge_vec, dist, sh1);

    for (int l = 0; l < 3; ++l) {
        hipMemsetAsync(agg_s, 0, (size_t)N_NODES * MUL * sizeof(float), stream);
        hipMemsetAsync(agg_v, 0, (size_t)N_NODES * MUL * 3 * sizeof(float), stream);
        edge_layer_kernel<<<E_EDGES / 64, 128, 0, stream>>>(src, dst, dist, sh1, s, v,
                                                            fc1[l], fc2p[l], fc3p[l],
                                                            agg_s, agg_v);
        // s = agg_s @ lin_s + s
        gemm64_kernel<<<N_NODES / 16, 128, 0, stream>>>(agg_s, 64, 1, lin_sp[l],
                                                        s, 64, 1, nullptr, 0,
                                                        s, 64, 1);
        // v[:, :, c] = agg_v[:, :, c] @ lin_v + v[:, :, c]
        for (int c = 0; c < 3; ++c)
            gemm64_kernel<<<N_NODES / 16, 128, 0, stream>>>(agg_v + c, 192, 3, lin_vp[l],
                                                            v + c, 192, 3, nullptr, 0,
                                                            v + c, 192, 3);
    }

    // onsite head
    gemm64_kernel<<<N_NODES / 16, 128, 0, stream>>>(s, 64, 1, ons_extp,
                                                    nullptr, 0, 0, nullptr, 0,
                                                    sc, 64, 1);
    gemm64_kernel<<<N_NODES / 16, 128, 0, stream>>>(sc, 64, 1, ons_w1p,
                                                    nullptr, 0, 0, ons_b1, 1,
                                                    t1, 64, 1);
    head4_kernel<<<(N_NODES * BD + 255) / 256, 256, 0, stream>>>(t1, ons_w2, ons_b2, onsite);

    lowrank_kernel<<<(N_NODES * 4 + 255) / 256, 256, 0, stream>>>(s, v, vecA_ws, vecA_wv,
                                                                  vecB_ws, vecB_wv, UA, UB);
    eigs_kernel<<<2 * KKEEP, 256, 0, stream>>>(onsite, UA, UB, evA, evB);
    final_kernel<<<1, 32, 0, stream>>>(evA, evB, en_w1, en_b1, en_w2, en_b2, (float*)d_out);
}